// Loss_52656299049080
// MI455X (gfx1250) — compile-verified
//
#include <hip/hip_runtime.h>
#include <math.h>

// ---------------------------------------------------------------------------
// Chamfer-distance loss for MI455X (gfx1250, wave32).
//
// d(a,b) = |a|^2 + |b|^2 - 2 a.b  folded into V_WMMA_F32_16X16X4_F32:
//   A row    = (x, y, z, |a|^2)            (K=4, lanes 0-15: K0,K1; 16-31: K2,K3)
//   B column = (-2bx, -2by, -2bz, 1)
//   C        = |b|^2 broadcast per column (C/D layout: lane%16 == column)
// => D = AxB + C is the 16x16 squared-distance tile, no post-fixup needed.
//
// Row-mins are kept in the 8 accumulator VGPRs across the full column sweep;
// the chamfer is run twice with swapped point sets so each direction's min is
// a register-resident row-min (no global atomic-min storm, one atomicAdd per
// wave). The 335M-entry distance matrix is never materialized (saves >1.3 GB
// of HBM traffic); inputs are ~1.8 MB and live in L2.
// ---------------------------------------------------------------------------

typedef __attribute__((ext_vector_type(2))) float v2f;
typedef __attribute__((ext_vector_type(8))) float v8f;

#define WAVES_PER_BLOCK 4

// Point idx of "concat(pa (na pts), pb (nb pts))" for batch b, layout (B, n, 3).
__device__ __forceinline__ void load_point(const float* __restrict__ pa, int na,
                                           const float* __restrict__ pb, int nb,
                                           int b, int idx,
                                           float& x, float& y, float& z) {
    const float* base;
    int local;
    if (idx < na) {
        base  = pa + (size_t)b * (size_t)na * 3u;
        local = idx;
    } else {
        base  = pb + (size_t)b * (size_t)nb * 3u;
        local = idx - na;
    }
    x = base[3 * local + 0];
    y = base[3 * local + 1];
    z = base[3 * local + 2];
}

__global__ void init_accum_kernel(float* acc) {
    acc[0] = 0.0f;
    acc[1] = 0.0f;
}

// One wave owns a 16-row tile of set A and sweeps all columns of set B,
// accumulating per-row mins, then wave-reduces and atomically adds the sum
// of its 16 row-min distances into *accum.
__global__ void __launch_bounds__(32 * WAVES_PER_BLOCK)
chamfer_rowmin_kernel(const float* __restrict__ rowsA, int nA,
                      const float* __restrict__ rowsA2, int nA2,
                      const float* __restrict__ colsB, int nB,
                      const float* __restrict__ colsB2, int nB2,
                      float* __restrict__ accum) {
    const int lane  = threadIdx.x & 31;
    const int wave  = threadIdx.x >> 5;
    const int b     = blockIdx.y;
    const int half  = lane >> 4;     // 0 => holds K=0,1 ; 1 => holds K=2,3
    const int li    = lane & 15;     // A: row-in-tile ; B: col-in-tile
    const int nCols = nB + nB2;

    const int row0 = (blockIdx.x * WAVES_PER_BLOCK + wave) * 16;

    // A fragment: point row0+li, folded as (x, y, z, |a|^2).
    float ax, ay, az;
    load_point(rowsA, nA, rowsA2, nA2, b, row0 + li, ax, ay, az);
    const float an2 = ax * ax + ay * ay + az * az;
    v2f afrag;
    afrag.x = half ? az  : ax;
    afrag.y = half ? an2 : ay;

    v8f rmin;
#pragma unroll
    for (int v = 0; v < 8; ++v) rmin[v] = __builtin_inff();

    for (int m0 = 0; m0 < nCols; m0 += 16) {
        // B fragment: column point m0+li, folded as (-2bx, -2by, -2bz, 1).
        float bx, by, bz;
        load_point(colsB, nB, colsB2, nB2, b, m0 + li, bx, by, bz);
        const float bn2 = bx * bx + by * by + bz * bz;
        v2f bfrag;
        bfrag.x = half ? (-2.0f * bz) : (-2.0f * bx);
        bfrag.y = half ? 1.0f         : (-2.0f * by);

        // C = |b|^2 broadcast (C/D layout: lane's column is li for all 8 regs).
        v8f c;
#pragma unroll
        for (int v = 0; v < 8; ++v) c[v] = bn2;

        // D[i][j] = |a_i|^2 - 2 a_i.b_j + |b_j|^2
        v8f d = __builtin_amdgcn_wmma_f32_16x16x4_f32(
            false, afrag, false, bfrag, (short)0, c, false, false);

#pragma unroll
        for (int v = 0; v < 8; ++v) rmin[v] = fminf(rmin[v], d[v]);
    }

    // Row v lives in lanes 0-15 (rows 0-7) / lanes 16-31 (rows 8-15):
    // xor-min over masks 1,2,4,8 reduces within each 16-lane half.
    float sum = 0.0f;
#pragma unroll
    for (int v = 0; v < 8; ++v) {
        float r = rmin[v];
#pragma unroll
        for (int off = 1; off <= 8; off <<= 1)
            r = fminf(r, __shfl_xor(r, off, 32));
        sum += r;   // uniform within each half
    }
    sum += __shfl_xor(sum, 16, 32);  // rows 0-7 mins + rows 8-15 mins
    if (lane == 0) atomicAdd(accum, sum);
}

// Pose loss (log-softmax over 16x64) + final weighted combine -> 5 outputs.
__global__ void finalize_kernel(const float* __restrict__ logits,
                                const float* __restrict__ acc,
                                float* __restrict__ out,
                                int B, int nH, int nRows1, int nCols1) {
    const int lane = threadIdx.x;  // single wave of 32
    float term = 0.0f;
    if (lane < B) {
        const float* l = logits + lane * nH;
        float mx = -__builtin_inff();
        for (int j = 0; j < nH; ++j) mx = fmaxf(mx, l[j]);
        float s = 0.0f;
        for (int j = 0; j < nH; ++j) s += __expf(l[j] - mx);
        const float lse = mx + __logf(s);
        term = l[0] - lse;  // log_prob[:,0]
    }
#pragma unroll
    for (int off = 16; off > 0; off >>= 1) term += __shfl_xor(term, off, 32);

    if (lane == 0) {
        const float pose  = -term / (float)B;
        const float shape = acc[0] / ((float)B * (float)nRows1)   // mean(dist1)
                          + acc[1] / ((float)B * (float)nCols1);  // mean(dist2)
        const float loss  = 0.4f * pose + 0.4f * shape + 0.1f * shape + 0.1f * shape;
        out[0] = loss;
        out[1] = pose;
        out[2] = shape;
        out[3] = shape;
        out[4] = shape;
    }
}

extern "C" void kernel_launch(void* const* d_in, const int* in_sizes, int n_in,
                              void* d_out, int out_size, void* d_ws, size_t ws_size,
                              hipStream_t stream) {
    (void)in_sizes; (void)n_in; (void)out_size; (void)ws_size;

    const float* shape_query = (const float*)d_in[0];  // (16, 4096, 3)
    const float* logits      = (const float*)d_in[1];  // (16, 64)
    // d_in[2] pair_caption, d_in[3] caption_embed, d_in[4] sdf_pred: unused
    const float* prior       = (const float*)d_in[5];  // (16, 1024, 3)
    const float* model       = (const float*)d_in[6];  // (16, 4096, 3)
    float* out = (float*)d_out;
    float* acc = (float*)d_ws;  // acc[0] = sum(dist1), acc[1] = sum(dist2)

    const int B = 16, Nq = 4096, Np = 1024, M = 4096;
    const int nRows1 = Nq + Np;  // 5120 rows = reconstructed
    const int rowsPerBlock = 16 * WAVES_PER_BLOCK;  // 64

    init_accum_kernel<<<1, 1, 0, stream>>>(acc);

    // dist1: rows = concat(shape_query, prior), cols = model
    chamfer_rowmin_kernel<<<dim3(nRows1 / rowsPerBlock, B),
                            32 * WAVES_PER_BLOCK, 0, stream>>>(
        shape_query, Nq, prior, Np, model, M, nullptr, 0, acc + 0);

    // dist2: rows = model, cols = concat(shape_query, prior)
    chamfer_rowmin_kernel<<<dim3(M / rowsPerBlock, B),
                            32 * WAVES_PER_BLOCK, 0, stream>>>(
        model, M, nullptr, 0, shape_query, Nq, prior, Np, acc + 1);

    finalize_kernel<<<1, 32, 0, stream>>>(logits, acc, out, B, 64, nRows1, M);
}